// SpectralAttentionBlock_34548716929432
// MI455X (gfx1250) — compile-verified
//
#include <hip/hip_runtime.h>
#include <hip/hip_bf16.h>
#include <math.h>

// SpectralAttentionBlock (channel attention) for gfx1250, fp32 WMMA path.
//
// Algebraic reformulation: with G_b = X_b^T X_b (64x64 Gram over n=16384 tokens),
//   S[b,h]   = Wk_h^T G_b Wq_h           (pre-norm logits)
//   ||k_d||^2 = (Wk_h^T G Wk_h)[d,d], ||q_e||^2 = (Wq_h^T G Wq_h)[e,e]
//   attn     = softmax_e( S[d,e] * rescale[h] / (max(||k_d||,eps)*max(||q_e||,eps)) )
//   out      = X_b @ W_eff_b + bp,  W_eff_b = Wv @ M_b,
//   M_b[h*64+e, c] = sum_d attn[b,h,d,e] * Wp[h*64+d, c]
// so the only O(n) passes are the Gram (read x once) and the final GEMM
// (read x once, write out once). q/k/v/o are never materialized.

#define B_   4
#define C_   64
#define N_   16384          // H*W
#define HEADS_ 8
#define INNER_ 512
#define EPS_ 1e-12f

typedef __attribute__((ext_vector_type(2))) float v2f;
typedef __attribute__((ext_vector_type(8))) float v8f;

static __device__ __forceinline__ v8f wmma_f32(v2f a, v2f b, v8f c) {
  // D[16x16] = A[16x4] * B[4x16] + C   (fp32 WMMA)
  return __builtin_amdgcn_wmma_f32_16x16x4_f32(false, a, false, b, (short)0, c,
                                               false, false);
}

// ---------------------------------------------------------------------------
// Kernel 1: per-block partial Gram matrices G_b = X_b^T X_b via f32 WMMA.
// Block = 64 threads (2 waves). Each wave covers 128 tokens, accumulating a
// full 64x64 partial in 16 accumulator tiles. Waves merge in LDS (fixed
// order, deterministic) and the block writes one 4096-float partial.
// ---------------------------------------------------------------------------
__global__ __launch_bounds__(64) void gram_kernel(const float* __restrict__ x,
                                                  float* __restrict__ partG) {
  const int b    = blockIdx.y;
  const int lane = threadIdx.x & 31;
  const int wave = threadIdx.x >> 5;
  const int half = lane >> 4;          // 0: K=0/1, 1: K=2/3
  const int l    = lane & 15;
  const float* X = x + (size_t)b * C_ * N_;
  const int n0   = (blockIdx.x * 2 + wave) * 128;

  v8f acc[16] = {};
  for (int kk = 0; kk < 32; ++kk) {
    const int tok = n0 + kk * 4 + 2 * half;
    v2f f[4];
#pragma unroll
    for (int g = 0; g < 4; ++g) {
      // A (and identically B) fragment: lane l holds X[ch, tok], X[ch, tok+1]
      f[g] = *(const v2f*)(X + (size_t)(g * 16 + l) * N_ + tok);
    }
#pragma unroll
    for (int gi = 0; gi < 4; ++gi)
#pragma unroll
      for (int gj = 0; gj < 4; ++gj)
        acc[gi * 4 + gj] = wmma_f32(f[gi], f[gj], acc[gi * 4 + gj]);
  }

  __shared__ float red[2][4096];
#pragma unroll
  for (int gi = 0; gi < 4; ++gi)
#pragma unroll
    for (int gj = 0; gj < 4; ++gj)
#pragma unroll
      for (int r = 0; r < 8; ++r) {
        const int d = gi * 16 + r + 8 * half;   // C/D layout: VGPR r -> M=r / r+8
        const int e = gj * 16 + l;              // N striped across lanes
        red[wave][d * 64 + e] = acc[gi * 4 + gj][r];
      }
  __syncthreads();

  float* outp = partG + ((size_t)b * 64 + blockIdx.x) * 4096;
  for (int i = threadIdx.x; i < 4096; i += 64)
    outp[i] = red[0][i] + red[1][i];
}

// ---------------------------------------------------------------------------
// Kernel 2: deterministic fixed-order reduction of Gram partials.
// ---------------------------------------------------------------------------
__global__ __launch_bounds__(256) void reduce_gram_kernel(
    const float* __restrict__ partG, float* __restrict__ G) {
  const int idx = blockIdx.x * 256 + threadIdx.x;   // 0 .. B*4096-1
  const int b = idx >> 12;
  const int i = idx & 4095;
  float s = 0.0f;
  for (int t = 0; t < 64; ++t)
    s += partG[((size_t)b * 64 + t) * 4096 + i];
  G[idx] = s;
}

// ---------------------------------------------------------------------------
// Kernel 3: per (b,h) attention. All 64x64 work in LDS. Writes the attn
// output (second tuple element) and a per-head W_eff partial.
// ---------------------------------------------------------------------------
__global__ __launch_bounds__(256) void attn_kernel(
    const float* __restrict__ G, const float* __restrict__ Wq,
    const float* __restrict__ Wk, const float* __restrict__ Wv,
    const float* __restrict__ Wp, const float* __restrict__ rescale,
    float* __restrict__ attn_out, float* __restrict__ partW) {
  const int h = blockIdx.x, b = blockIdx.y;
  const int tid = threadIdx.x;
  __shared__ float sG[4096], b1[4096], b2[4096];
  __shared__ float sQ[64], sK[64];

  for (int i = tid; i < 4096; i += 256) sG[i] = G[b * 4096 + i];
  __syncthreads();

  // b1 = G @ Wq_h ; b2 = G @ Wk_h
  for (int i = tid; i < 4096; i += 256) {
    const int c = i >> 6, col = i & 63;
    float s1 = 0.0f, s2 = 0.0f;
    for (int k = 0; k < 64; ++k) {
      const float g = sG[c * 64 + k];
      s1 += g * Wq[k * INNER_ + h * 64 + col];
      s2 += g * Wk[k * INNER_ + h * 64 + col];
    }
    b1[i] = s1;
    b2[i] = s2;
  }
  __syncthreads();

  // inverse norms from diagonals of Wq^T G Wq and Wk^T G Wk
  if (tid < 64) {
    const int e = tid;
    float qd = 0.0f, kd = 0.0f;
    for (int c = 0; c < 64; ++c) {
      qd += Wq[c * INNER_ + h * 64 + e] * b1[c * 64 + e];
      kd += Wk[c * INNER_ + h * 64 + e] * b2[c * 64 + e];
    }
    sQ[e] = 1.0f / fmaxf(sqrtf(fmaxf(qd, 0.0f)), EPS_);
    sK[e] = 1.0f / fmaxf(sqrtf(fmaxf(kd, 0.0f)), EPS_);
  }
  __syncthreads();

  // P = Wk_h^T (G Wq_h)  -> b2 (T2 consumed)
  for (int i = tid; i < 4096; i += 256) {
    const int d = i >> 6, e = i & 63;
    float s = 0.0f;
    for (int c = 0; c < 64; ++c)
      s += Wk[c * INNER_ + h * 64 + d] * b1[c * 64 + e];
    b2[i] = s;
  }
  __syncthreads();

  // row softmax -> attn into b1, also emitted as second tuple output
  const float rsc = rescale[h];
  if (tid < 64) {
    const int d = tid;
    const float invk = sK[d];
    float m = -3.4e38f;
    for (int e = 0; e < 64; ++e)
      m = fmaxf(m, b2[d * 64 + e] * rsc * invk * sQ[e]);
    float sum = 0.0f;
    for (int e = 0; e < 64; ++e) {
      const float v = expf(b2[d * 64 + e] * rsc * invk * sQ[e] - m);
      b1[d * 64 + e] = v;
      sum += v;
    }
    const float inv = 1.0f / sum;
    float* ao = attn_out + (((size_t)b * HEADS_ + h) * 64 + d) * 64;
    for (int e = 0; e < 64; ++e) {
      const float a = b1[d * 64 + e] * inv;
      b1[d * 64 + e] = a;
      ao[e] = a;
    }
  }
  __syncthreads();

  // M[e][c] = sum_d attn[d][e] * Wp[h*64+d, c]  -> b2 (P consumed)
  for (int i = tid; i < 4096; i += 256) {
    const int e = i >> 6, c = i & 63;
    float s = 0.0f;
    for (int d = 0; d < 64; ++d)
      s += b1[d * 64 + e] * Wp[(h * 64 + d) * 64 + c];
    b2[i] = s;
  }
  __syncthreads();

  // per-head W_eff partial: W[cin][cout] = sum_e Wv[cin, h*64+e] * M[e][cout]
  const int cin = tid >> 2;
  const int cout0 = (tid & 3) * 16;
  float w[16] = {};
  for (int e = 0; e < 64; ++e) {
    const float wv = Wv[cin * INNER_ + h * 64 + e];
    const float* mrow = &b2[e * 64 + cout0];
#pragma unroll
    for (int j = 0; j < 16; ++j) w[j] += wv * mrow[j];
  }
  float* op = partW + ((size_t)(b * HEADS_ + h)) * 4096 + tid * 16;
#pragma unroll
  for (int j = 0; j < 16; ++j) op[j] = w[j];
}

// ---------------------------------------------------------------------------
// Kernel 4: deterministic reduction of W_eff over heads.
// ---------------------------------------------------------------------------
__global__ __launch_bounds__(256) void reduce_weff_kernel(
    const float* __restrict__ partW, float* __restrict__ Weff) {
  const int idx = blockIdx.x * 256 + threadIdx.x;   // 0 .. B*4096-1
  const int b = idx >> 12;
  const int i = idx & 4095;
  float s = 0.0f;
  for (int h = 0; h < HEADS_; ++h)
    s += partW[((size_t)(b * HEADS_ + h)) * 4096 + i];
  Weff[idx] = s;
}

// ---------------------------------------------------------------------------
// Kernel 5: out[b,c,n] = (W_eff_b^T X_b)[c,n] + bp[c], f32 WMMA.
// Block = 256 threads (8 waves); each wave owns a 16-token strip, 64 output
// channels (4 M-tiles), K-loop over 64 input channels in steps of 4.
// ---------------------------------------------------------------------------
__global__ __launch_bounds__(256) void out_kernel(const float* __restrict__ x,
                                                  const float* __restrict__ Weff,
                                                  const float* __restrict__ bp,
                                                  float* __restrict__ out) {
  const int b = blockIdx.y;
  __shared__ float sW[4096];
  __shared__ float sB[64];
  for (int i = threadIdx.x; i < 4096; i += 256) sW[i] = Weff[b * 4096 + i];
  if (threadIdx.x < 64) sB[threadIdx.x] = bp[threadIdx.x];
  __syncthreads();

  const int lane = threadIdx.x & 31;
  const int wave = threadIdx.x >> 5;
  const int half = lane >> 4;
  const int l    = lane & 15;
  const int n0   = blockIdx.x * 128 + wave * 16;
  const float* X = x + (size_t)b * C_ * N_;

  v8f acc[4] = {};
  for (int k0 = 0; k0 < 64; k0 += 4) {
    const int ch = k0 + 2 * half;
    v2f bv;                                  // B[k, n] = x[ch, n0+n]
    bv.x = X[(size_t)ch * N_ + n0 + l];
    bv.y = X[(size_t)(ch + 1) * N_ + n0 + l];
#pragma unroll
    for (int mi = 0; mi < 4; ++mi) {
      v2f av;                                // A[m, k] = Weff[ch, mi*16+m]
      av.x = sW[ch * 64 + mi * 16 + l];
      av.y = sW[(ch + 1) * 64 + mi * 16 + l];
      acc[mi] = wmma_f32(av, bv, acc[mi]);
    }
  }

  float* O = out + (size_t)b * C_ * N_;
#pragma unroll
  for (int mi = 0; mi < 4; ++mi)
#pragma unroll
    for (int r = 0; r < 8; ++r) {
      const int c = mi * 16 + r + 8 * half;
      O[(size_t)c * N_ + n0 + l] = acc[mi][r] + sB[c];
    }
}

// ---------------------------------------------------------------------------
extern "C" void kernel_launch(void* const* d_in, const int* in_sizes, int n_in,
                              void* d_out, int out_size, void* d_ws,
                              size_t ws_size, hipStream_t stream) {
  const float* x       = (const float*)d_in[0];  // [B, C, H, W]
  const float* Wq      = (const float*)d_in[1];  // [C, INNER]
  const float* Wk      = (const float*)d_in[2];
  const float* Wv      = (const float*)d_in[3];
  const float* Wp      = (const float*)d_in[4];  // [INNER, C]
  const float* bp      = (const float*)d_in[5];  // [C]
  const float* rescale = (const float*)d_in[6];  // [HEADS]

  float* out      = (float*)d_out;                       // [B, C, H, W]
  float* attn_out = out + (size_t)B_ * C_ * N_;          // [B, HEADS, 64, 64]

  // workspace layout (floats)
  float* ws    = (float*)d_ws;
  float* partG = ws;                       // B * 64 tiles * 4096 = 1,048,576
  float* G     = partG + (size_t)B_ * 64 * 4096;      // B * 4096
  float* partW = G + (size_t)B_ * 4096;               // B * HEADS * 4096
  float* Weff  = partW + (size_t)B_ * HEADS_ * 4096;  // B * 4096

  (void)in_sizes; (void)n_in; (void)out_size; (void)ws_size;

  gram_kernel<<<dim3(64, B_), 64, 0, stream>>>(x, partG);
  reduce_gram_kernel<<<(B_ * 4096) / 256, 256, 0, stream>>>(partG, G);
  attn_kernel<<<dim3(HEADS_, B_), 256, 0, stream>>>(G, Wq, Wk, Wv, Wp, rescale,
                                                    attn_out, partW);
  reduce_weff_kernel<<<(B_ * 4096) / 256, 256, 0, stream>>>(partW, Weff);
  out_kernel<<<dim3(N_ / 128, B_), 256, 0, stream>>>(x, Weff, bp, out);
}